// SquareDiagonalLinear_79723182948537
// MI455X (gfx1250) — compile-verified
//
#include <hip/hip_runtime.h>

// out[r, c] = x[r, c] * w + bias[c], x: 16384 x 4096 f32.
// Memory-bound streaming op: 512 MiB compulsory HBM traffic -> ~23 us floor
// @ 23.3 TB/s. 134 MFLOP is negligible; WMMA is the wrong tool (AI = 0.25
// FLOP/byte). Optimal mix: 128-bit non-temporal global loads/stores (CDNA5
// TH cache controls), bias held in registers, wave32-friendly blocks.

typedef float f4 __attribute__((ext_vector_type(4)));

// ---------------------------------------------------------------------------
// Fast path: requires ncols4 == 4 * blockDim.x (4096 f32 columns, 256 thr).
// Each thread owns 4 fixed float4 column slots; bias loaded ONCE into
// registers, then rows are streamed with exactly 4 NT loads + 4 FMA4 +
// 4 NT stores per row.
// ---------------------------------------------------------------------------
__global__ __launch_bounds__(256) void scale_bias_rows_kernel(
    const f4* __restrict__ x4,
    const float* __restrict__ wptr,
    const f4* __restrict__ b4,
    f4* __restrict__ out4,
    unsigned nrows,
    unsigned ncols4)   // float4 columns per row; == 4 * blockDim.x
{
    const float w = wptr[0];                    // wave-uniform, s_load + cache
    const unsigned t  = threadIdx.x;
    const unsigned bd = blockDim.x;
    const unsigned c0 = t;
    const unsigned c1 = t + bd;
    const unsigned c2 = t + 2u * bd;
    const unsigned c3 = t + 3u * bd;

    // Bias: 4 vectors per thread, loaded once, reused for every row.
    const f4 bb0 = b4[c0];
    const f4 bb1 = b4[c1];
    const f4 bb2 = b4[c2];
    const f4 bb3 = b4[c3];

    const unsigned row_step  = gridDim.x;
    const unsigned base_step = row_step * ncols4;
    unsigned base = blockIdx.x * ncols4;

    for (unsigned r = blockIdx.x; r < nrows; r += row_step, base += base_step) {
        f4 a0 = __builtin_nontemporal_load(&x4[base + c0]);
        f4 a1 = __builtin_nontemporal_load(&x4[base + c1]);
        f4 a2 = __builtin_nontemporal_load(&x4[base + c2]);
        f4 a3 = __builtin_nontemporal_load(&x4[base + c3]);

        f4 r0 = a0 * w + bb0;
        f4 r1 = a1 * w + bb1;
        f4 r2 = a2 * w + bb2;
        f4 r3 = a3 * w + bb3;

        __builtin_nontemporal_store(r0, &out4[base + c0]);
        __builtin_nontemporal_store(r1, &out4[base + c1]);
        __builtin_nontemporal_store(r2, &out4[base + c2]);
        __builtin_nontemporal_store(r3, &out4[base + c3]);
    }
}

// ---------------------------------------------------------------------------
// Generic fallback: grid-stride, 4-deep unroll, masked bias indexing
// (bias_n4 power of two), scalar tail for sizes not divisible by 4.
// ---------------------------------------------------------------------------
__global__ __launch_bounds__(256) void scale_bias_generic_kernel(
    const f4* __restrict__ x4,
    const float* __restrict__ wptr,
    const f4* __restrict__ b4,
    f4* __restrict__ out4,
    const float* __restrict__ x_s,
    const float* __restrict__ b_s,
    float* __restrict__ out_s,
    unsigned n4,
    unsigned n_total,
    unsigned bias_mask4,
    unsigned bias_len)
{
    const float w = wptr[0];
    const unsigned stride = gridDim.x * blockDim.x;
    unsigned i = blockIdx.x * blockDim.x + threadIdx.x;

    for (; i + 3u * stride < n4; i += 4u * stride) {
        const unsigned i0 = i;
        const unsigned i1 = i + stride;
        const unsigned i2 = i + 2u * stride;
        const unsigned i3 = i + 3u * stride;

        f4 a0 = __builtin_nontemporal_load(&x4[i0]);
        f4 a1 = __builtin_nontemporal_load(&x4[i1]);
        f4 a2 = __builtin_nontemporal_load(&x4[i2]);
        f4 a3 = __builtin_nontemporal_load(&x4[i3]);

        f4 c0 = b4[i0 & bias_mask4];
        f4 c1 = b4[i1 & bias_mask4];
        f4 c2 = b4[i2 & bias_mask4];
        f4 c3 = b4[i3 & bias_mask4];

        __builtin_nontemporal_store(a0 * w + c0, &out4[i0]);
        __builtin_nontemporal_store(a1 * w + c1, &out4[i1]);
        __builtin_nontemporal_store(a2 * w + c2, &out4[i2]);
        __builtin_nontemporal_store(a3 * w + c3, &out4[i3]);
    }

    for (; i < n4; i += stride) {
        f4 a = __builtin_nontemporal_load(&x4[i]);
        f4 c = b4[i & bias_mask4];
        __builtin_nontemporal_store(a * w + c, &out4[i]);
    }

    if (blockIdx.x == 0 && threadIdx.x == 0) {
        for (unsigned k = n4 * 4u; k < n_total; ++k) {
            out_s[k] = x_s[k] * w + b_s[k % bias_len];
        }
    }
}

extern "C" void kernel_launch(void* const* d_in, const int* in_sizes, int n_in,
                              void* d_out, int out_size, void* d_ws, size_t ws_size,
                              hipStream_t stream) {
    const float* x    = (const float*)d_in[0];  // (16384, 4096) f32
    const float* w    = (const float*)d_in[1];  // (1,) f32
    const float* bias = (const float*)d_in[2];  // (4096,) f32
    float* out        = (float*)d_out;

    const unsigned n_total  = (unsigned)out_size;     // 67,108,864
    const unsigned bias_len = (unsigned)in_sizes[2];  // 4096
    const unsigned threads  = 256;                    // 8 wave32 waves / block

    const bool fast =
        (bias_len % 4u == 0u) &&
        ((bias_len / 4u) == 4u * threads) &&           // ncols4 == 4*blockDim
        (n_total % bias_len == 0u);

    if (fast) {
        const unsigned ncols4 = bias_len / 4u;         // 1024
        const unsigned nrows  = n_total / bias_len;    // 16384
        unsigned blocks = nrows < 4096u ? nrows : 4096u; // 4 rows per block
        if (blocks == 0) blocks = 1;
        scale_bias_rows_kernel<<<blocks, threads, 0, stream>>>(
            (const f4*)x, w, (const f4*)bias, (f4*)out, nrows, ncols4);
    } else {
        const unsigned n4 = n_total / 4u;
        unsigned bias_n4 = bias_len / 4u;
        if (bias_n4 == 0) bias_n4 = 1;
        const unsigned bias_mask4 = bias_n4 - 1u;      // bias_n4 assumed pow2
        const unsigned unroll = 4;
        unsigned blocks = (n4 + threads * unroll - 1u) / (threads * unroll);
        if (blocks == 0) blocks = 1;
        scale_bias_generic_kernel<<<blocks, threads, 0, stream>>>(
            (const f4*)x, w, (const f4*)bias, (f4*)out,
            x, bias, out, n4, n_total, bias_mask4, bias_len);
    }
}